// SmoothnessLoss_23046794510778
// MI455X (gfx1250) — compile-verified
//
#include <hip/hip_runtime.h>
#include <hip/hip_bf16.h>

// Problem constants (match reference)
#define HW        224
#define NPIX      (HW * HW)        // 50176 = 196 * 256 exactly
#define KK        441              // 21*21 offsets
#define QCHUNKS   8
#define QPER      56               // ceil(441/8)
#define NBLK_PIX  196
#define NPART     (NBLK_PIX * QCHUNKS)   // 1568 partials -> d_ws

#define SIGMA_COLOR 50.0f                 // 1/(0.1^2 * 2)
#define SIGMA_SPACE 0.010204081632653061f // 1/(7^2 * 2)
#define C1_THRESH   20.0f
#define C2_THRESH   10.0f

typedef float v2f __attribute__((ext_vector_type(2)));
typedef float v8f __attribute__((ext_vector_type(8)));

__device__ __forceinline__ float ldpad(const float* __restrict__ img, int c, int y, int x) {
    bool ok = ((unsigned)y < (unsigned)HW) && ((unsigned)x < (unsigned)HW);
    return ok ? img[c * NPIX + y * HW + x] : 0.0f;
}

// Sobel gradient magnitude (cross-correlation, SAME zero pad) summed over 3 channels.
// Magnitude is invariant to kernel flip, so conv-vs-xcorr ambiguity is moot.
__device__ float edge_resp(const float* __restrict__ img, int y, int x) {
    float s = 0.0f;
#pragma unroll
    for (int c = 0; c < 3; ++c) {
        float a00 = ldpad(img, c, y - 1, x - 1);
        float a01 = ldpad(img, c, y - 1, x    );
        float a02 = ldpad(img, c, y - 1, x + 1);
        float a10 = ldpad(img, c, y,     x - 1);
        float a12 = ldpad(img, c, y,     x + 1);
        float a20 = ldpad(img, c, y + 1, x - 1);
        float a21 = ldpad(img, c, y + 1, x    );
        float a22 = ldpad(img, c, y + 1, x + 1);
        float gx = (a02 - a00) + 2.0f * (a12 - a10) + (a22 - a20);
        float gy = (a20 - a00) + 2.0f * (a21 - a01) + (a22 - a02);
        s += sqrtf(gx * gx + gy * gy + 1e-12f);
    }
    return s;
}

__global__ __launch_bounds__(256) void smooth_loss_main(
    const float* __restrict__ orig, const float* __restrict__ smo,
    float* __restrict__ part) {
    __shared__ float etab[21];   // separable spatial weight table
    __shared__ float red[256];   // block reduction staging

    const int tid = threadIdx.x;
    if (tid < 21) {
        float d = (float)(tid - 10);
        etab[tid] = __expf(-SIGMA_SPACE * d * d);
    }
    __syncthreads();

    const int p  = blockIdx.x * 256 + tid;   // pixel id, exact cover of 50176
    const int yp = p / HW;
    const int xp = p - yp * HW;

    // Center values: index c*NPIX + yp*224 + xp == c*NPIX + p
    const float scn[3] = { smo[p],  smo[NPIX + p],  smo[2 * NPIX + p]  };
    const float ocn[3] = { orig[p], orig[NPIX + p], orig[2 * NPIX + p] };

    const float ero = edge_resp(orig, yp, xp);
    const float ers = edge_resp(smo,  yp, xp);
    const bool  uL  = (ero < C1_THRESH) && ((ers - ero) > C2_THRESH);

    const int q0 = blockIdx.y * QPER;
    const int q1 = (q0 + QPER < KK) ? (q0 + QPER) : KK;

    float acc = 0.0f;
    for (int q = q0; q < q1; ++q) {
        const unsigned iq = (unsigned)q / 21u;
        const unsigned jq = (unsigned)q - 21u * iq;
        const float wsq = etab[iq] * etab[jq];

        float t2sum = 0.0f, cd = 0.0f, psum = 0.0f;
#pragma unroll
        for (int cp = 0; cp < 3; ++cp) {
            // Decode the row-major reshape scramble: view (q,cp,yp,xp) ->
            // patch array index [c][y][x][kh][kw]
            const unsigned u  = 3u * (unsigned)q + (unsigned)cp;
            const unsigned c  = u / 441u;
            const unsigned v  = u - 441u * c;
            const unsigned A  = v * 224u + (unsigned)yp;
            const unsigned yy = A / 441u;
            const unsigned Bv = A - 441u * yy;
            const unsigned Cc = Bv * 224u + (unsigned)xp;
            const unsigned xx = Cc / 441u;
            const unsigned r3 = Cc - 441u * xx;
            const unsigned kh = r3 / 21u;
            const unsigned kw = r3 - 21u * kh;

            const int sy = (int)yy + (int)kh - 10;
            const int sx = (int)xx + (int)kw - 10;
            const bool ok = ((unsigned)sy < (unsigned)HW) && ((unsigned)sx < (unsigned)HW);
            const int off = (int)c * NPIX + sy * HW + sx;

            const float sp = ok ? smo[off]  : 0.0f;   // zero pad == (sp>0) masked out
            const float op = ok ? orig[off] : 0.0f;

            const float tij = (sp > 0.0f) ? fabsf(sp - scn[cp]) : 0.0f;
            const float eo  = op - ocn[cp];
            cd += (op > 0.0f) ? eo * eo : 0.0f;

            if (uL) {
                t2sum += tij * tij;
            } else {
                // safe_pow(x, 0.8): hardware v_log_f32 + v_exp_f32
                psum += (tij > 0.0f) ? exp2f(0.8f * __log2f(tij)) : 0.0f;
            }
        }
        acc += uL ? (wsq * t2sum) : (__expf(-SIGMA_COLOR * cd) * psum);
    }

    // ---- Block reduction: 256 partials via chained V_WMMA_F32_16X16X4_F32 ----
    // A = ones(16x4) f32  =>  D[m][n] = sum_k B[k][n] + C[m][n]; 4 chained WMMAs
    // consume all 256 values with exact f32 RNE accumulation.
    red[tid] = acc;
    __syncthreads();
    if (tid < 32) {               // wave 0, fully active => EXEC all 1s
        v2f a;  a[0] = 1.0f; a[1] = 1.0f;
        v8f cacc = {0.0f, 0.0f, 0.0f, 0.0f, 0.0f, 0.0f, 0.0f, 0.0f};
#pragma unroll
        for (int j = 0; j < 4; ++j) {
            v2f b;
            b[0] = red[tid * 8 + 2 * j];
            b[1] = red[tid * 8 + 2 * j + 1];
            cacc = __builtin_amdgcn_wmma_f32_16x16x4_f32(
                false, a, false, b, (short)0, cacc, false, false);
        }
        // Row 0 (lanes 0-15 of VGPR0) holds the 16 column sums; lanes 16-31 hold
        // the identical M=8 row. Fold across 32 lanes and halve.
        float vsum = cacc[0];
#pragma unroll
        for (int off = 16; off > 0; off >>= 1)
            vsum += __shfl_down(vsum, off, 32);
        if (tid == 0)
            part[blockIdx.y * NBLK_PIX + blockIdx.x] = 0.5f * vsum;
    }
}

// Deterministic final reduction (fixed order, no atomics -> bit-identical replays)
__global__ __launch_bounds__(256) void smooth_loss_final(
    const float* __restrict__ part, float* __restrict__ out) {
    __shared__ float s[256];
    const int t = threadIdx.x;
    float v = 0.0f;
    for (int i = t; i < NPART; i += 256) v += part[i];
    s[t] = v;
    __syncthreads();
    for (int k = 128; k > 0; k >>= 1) {
        if (t < k) s[t] += s[t + k];
        __syncthreads();
    }
    if (t == 0) out[0] = s[0] * (1.0f / (float)NPIX);
}

extern "C" void kernel_launch(void* const* d_in, const int* in_sizes, int n_in,
                              void* d_out, int out_size, void* d_ws, size_t ws_size,
                              hipStream_t stream) {
    (void)in_sizes; (void)n_in; (void)out_size; (void)ws_size;
    const float* orig = (const float*)d_in[0];   // original_images (1,3,224,224) f32
    const float* smo  = (const float*)d_in[1];   // smooth_images   (1,3,224,224) f32
    float* part = (float*)d_ws;                  // 1568 f32 partials (6.3 KB)

    dim3 grid(NBLK_PIX, QCHUNKS);
    smooth_loss_main<<<grid, 256, 0, stream>>>(orig, smo, part);
    smooth_loss_final<<<1, 256, 0, stream>>>(part, (float*)d_out);
}